// RecurrentGCN_77799037599894
// MI455X (gfx1250) — compile-verified
//
#include <hip/hip_runtime.h>

typedef __attribute__((ext_vector_type(2))) float v2f;
typedef __attribute__((ext_vector_type(8))) float v8f;

#define F 32                 // feature width (F_IN == F_H == 32)
#define TPB 256              // threads per block (8 wave32)
#define WAVES_PB 8
#define BSTRIDE 96           // floats per swizzled pair-row (96 % 64 == 32 -> no bank conflicts)
#define WSZ (16 * BSTRIDE)   // floats per swizzled 32x32 weight matrix

// ---------------------------------------------------------------------------
// fp32 WMMA wrapper: D = A(16x4) * B(4x16) + C  (V_WMMA_F32_16X16X4_F32)
// ---------------------------------------------------------------------------
__device__ __forceinline__ v8f wmma4(v2f a, v2f b, v8f c) {
  return __builtin_amdgcn_wmma_f32_16x16x4_f32(
      /*neg_a=*/false, a, /*neg_b=*/false, b,
      /*c_mod=*/(short)0, c, /*reuse_a=*/false, /*reuse_b=*/false);
}

// A fragment (16x4 tile of a row-major NxF matrix starting at row0, cols k0..k0+3)
// lanes 0-15: M=lane, K={k0,k0+1}; lanes 16-31: M=lane-16, K={k0+2,k0+3}
__device__ __forceinline__ v2f afrag(const float* __restrict__ M, int row0,
                                     int lane, int k0, int N) {
  int r = row0 + (lane & 15);
  r = r < N ? r : N - 1;
  int kk = k0 + ((lane >> 4) << 1);
  const float* p = M + (size_t)r * F + kk;
  v2f a; a.x = p[0]; a.y = p[1];
  return a;
}

// B fragment from swizzled LDS weight: pair-row p holds (W[2p][n], W[2p+1][n])
// contiguous at p*BSTRIDE + 2n  ->  one aligned ds_load_b64 per fragment.
__device__ __forceinline__ v2f bfragS(const float* __restrict__ W, int lane,
                                      int k0, int n0) {
  int p = (k0 >> 1) + (lane >> 4);     // lanes 0-15: rows {k0,k0+1}; 16-31: {k0+2,k0+3}
  int n = n0 + (lane & 15);
  return *(const v2f*)(W + p * BSTRIDE + 2 * n);
}

// stage one ChebConv weight tensor (2,32,32) into two swizzled LDS blocks
__device__ __forceinline__ void stage_w(float* __restrict__ dst,
                                        const float* __restrict__ src) {
  for (int idx = threadIdx.x; idx < 2 * F * F; idx += TPB) {
    int mat = idx >> 10;               // W[0] or W[1]
    int rem = idx & 1023;
    int kk = rem >> 5, n = rem & 31;
    dst[mat * WSZ + (kk >> 1) * BSTRIDE + 2 * n + (kk & 1)] = src[idx];
  }
}

__device__ __forceinline__ float sigmoidf_(float v) {
  return 1.0f / (1.0f + __expf(-v));
}

// ---------------------------------------------------------------------------
// scratch zeroing
// ---------------------------------------------------------------------------
__global__ void k_zero(float* __restrict__ p, long long n) {
  long long i = (long long)blockIdx.x * blockDim.x + threadIdx.x;
  long long stride = (long long)gridDim.x * blockDim.x;
  for (; i < n; i += stride) p[i] = 0.0f;
}

// deg[src] += w (self-loops zeroed)
__global__ void k_degree(const long long* __restrict__ ei,
                         const float* __restrict__ ew,
                         float* __restrict__ deg, int E) {
  int e = blockIdx.x * blockDim.x + threadIdx.x;
  if (e >= E) return;
  long long s = ei[e], d = ei[(size_t)E + e];
  float w = (s == d) ? 0.0f : ew[e];
  if (w != 0.0f) unsafeAtomicAdd(&deg[s], w);
}

// dinv in place over deg
__global__ void k_dinv(float* __restrict__ deg, int N) {
  int i = blockIdx.x * blockDim.x + threadIdx.x;
  if (i >= N) return;
  float dg = deg[i];
  deg[i] = (dg > 0.0f) ? rsqrtf(fmaxf(dg, 1e-30f)) : 0.0f;
}

// nw[e] = dinv[src] * w * dinv[dst]
__global__ void k_nw(const long long* __restrict__ ei,
                     const float* __restrict__ ew,
                     const float* __restrict__ dinv,
                     float* __restrict__ nw, int E) {
  int e = blockIdx.x * blockDim.x + threadIdx.x;
  if (e >= E) return;
  long long s = ei[e], d = ei[(size_t)E + e];
  float w = (s == d) ? 0.0f : ew[e];
  nw[e] = dinv[s] * w * dinv[d];
}

// fused SpMM for two node matrices: agg* accumulates Tx1 = -sum(nw * M[src]) at dst.
// one wave32 per edge; lane = feature.
__global__ void k_spmm2(const long long* __restrict__ ei,
                        const float* __restrict__ nw,
                        const float* __restrict__ A,
                        const float* __restrict__ B,
                        float* __restrict__ aggA,
                        float* __restrict__ aggB, int E) {
  int e = blockIdx.x * WAVES_PB + (threadIdx.x >> 5);
  if (e >= E) return;
  int lane = threadIdx.x & 31;
  float w = nw[e];
  if (w == 0.0f) return;
  long long s = ei[e], d = ei[(size_t)E + e];
  size_t so = (size_t)s * F + lane, dofs = (size_t)d * F + lane;
  unsafeAtomicAdd(&aggA[dofs], -w * A[so]);
  unsafeAtomicAdd(&aggB[dofs], -w * B[so]);
}

__global__ void k_spmm1(const long long* __restrict__ ei,
                        const float* __restrict__ nw,
                        const float* __restrict__ A,
                        float* __restrict__ aggA, int E) {
  int e = blockIdx.x * WAVES_PB + (threadIdx.x >> 5);
  if (e >= E) return;
  int lane = threadIdx.x & 31;
  float w = nw[e];
  if (w == 0.0f) return;
  long long s = ei[e], d = ei[(size_t)E + e];
  unsafeAtomicAdd(&aggA[(size_t)d * F + lane], -w * A[(size_t)s * F + lane]);
}

// ---------------------------------------------------------------------------
// GATE: Z = sigmoid(x@Wxz0 + Tx1x@Wxz1 + H@Whz0 + Tx1H@Whz1 + bz)
//       R = sigmoid(... Wxr/Whr ...);  HR = H * R
// one wave per 16-row tile; 128 fp32 WMMAs per wave.
// ---------------------------------------------------------------------------
__global__ void __launch_bounds__(TPB)
k_gate(const float* __restrict__ x, const float* __restrict__ H,
       const float* __restrict__ ax, const float* __restrict__ aH,
       const float* __restrict__ Wxz, const float* __restrict__ bxz,
       const float* __restrict__ Whz, const float* __restrict__ bhz,
       const float* __restrict__ Wxr, const float* __restrict__ bxr,
       const float* __restrict__ Whr, const float* __restrict__ bhr,
       float* __restrict__ Z, float* __restrict__ HR, int N) {
  __shared__ __align__(16) float sWxz[2 * WSZ], sWhz[2 * WSZ];
  __shared__ __align__(16) float sWxr[2 * WSZ], sWhr[2 * WSZ];
  __shared__ float sbz[F], sbr[F];
  stage_w(sWxz, Wxz);
  stage_w(sWhz, Whz);
  stage_w(sWxr, Wxr);
  stage_w(sWhr, Whr);
  if (threadIdx.x < F) {
    sbz[threadIdx.x] = bxz[threadIdx.x] + bhz[threadIdx.x];
    sbr[threadIdx.x] = bxr[threadIdx.x] + bhr[threadIdx.x];
  }
  __syncthreads();

  int lane = threadIdx.x & 31;
  int tile = blockIdx.x * WAVES_PB + (threadIdx.x >> 5);
  int row0 = tile * 16;
  if (row0 >= N) return;   // whole wave exits -> EXEC all ones at every WMMA

  int nlo = lane & 15;
  float bz0 = sbz[nlo], bz1 = sbz[16 + nlo];
  float br0 = sbr[nlo], br1 = sbr[16 + nlo];
  v8f z0 = {bz0, bz0, bz0, bz0, bz0, bz0, bz0, bz0};
  v8f z1 = {bz1, bz1, bz1, bz1, bz1, bz1, bz1, bz1};
  v8f r0 = {br0, br0, br0, br0, br0, br0, br0, br0};
  v8f r1 = {br1, br1, br1, br1, br1, br1, br1, br1};

#pragma unroll
  for (int k0 = 0; k0 < F; k0 += 4) {
    v2f fx  = afrag(x,  row0, lane, k0, N);
    v2f fax = afrag(ax, row0, lane, k0, N);
    v2f fh  = afrag(H,  row0, lane, k0, N);
    v2f faH = afrag(aH, row0, lane, k0, N);
    // Z gate
    z0 = wmma4(fx,  bfragS(sWxz,       lane, k0, 0),  z0);
    z1 = wmma4(fx,  bfragS(sWxz,       lane, k0, 16), z1);
    z0 = wmma4(fax, bfragS(sWxz + WSZ, lane, k0, 0),  z0);
    z1 = wmma4(fax, bfragS(sWxz + WSZ, lane, k0, 16), z1);
    z0 = wmma4(fh,  bfragS(sWhz,       lane, k0, 0),  z0);
    z1 = wmma4(fh,  bfragS(sWhz,       lane, k0, 16), z1);
    z0 = wmma4(faH, bfragS(sWhz + WSZ, lane, k0, 0),  z0);
    z1 = wmma4(faH, bfragS(sWhz + WSZ, lane, k0, 16), z1);
    // R gate
    r0 = wmma4(fx,  bfragS(sWxr,       lane, k0, 0),  r0);
    r1 = wmma4(fx,  bfragS(sWxr,       lane, k0, 16), r1);
    r0 = wmma4(fax, bfragS(sWxr + WSZ, lane, k0, 0),  r0);
    r1 = wmma4(fax, bfragS(sWxr + WSZ, lane, k0, 16), r1);
    r0 = wmma4(fh,  bfragS(sWhr,       lane, k0, 0),  r0);
    r1 = wmma4(fh,  bfragS(sWhr,       lane, k0, 16), r1);
    r0 = wmma4(faH, bfragS(sWhr + WSZ, lane, k0, 0),  r0);
    r1 = wmma4(faH, bfragS(sWhr + WSZ, lane, k0, 16), r1);
  }

  // D layout: VGPR r -> row (row0 + r + 8*(lane>=16)), col (lane&15) + n0
  int rbase = row0 + ((lane >> 4) << 3);
#pragma unroll
  for (int r = 0; r < 8; ++r) {
    int m = rbase + r;
    if (m >= N) break;
    size_t o0 = (size_t)m * F + nlo, o1 = o0 + 16;
    float zv0 = sigmoidf_(z0[r]), zv1 = sigmoidf_(z1[r]);
    float rv0 = sigmoidf_(r0[r]), rv1 = sigmoidf_(r1[r]);
    Z[o0] = zv0;             Z[o1] = zv1;
    HR[o0] = rv0 * H[o0];    HR[o1] = rv1 * H[o1];
  }
}

// ---------------------------------------------------------------------------
// CAND: Ht = tanh(x@Wxh0 + Tx1x@Wxh1 + HR@Whh0 + Tx1HR@Whh1 + bh)
//       h  = Z*H + (1-Z)*Ht    (64 WMMAs per wave)
// ---------------------------------------------------------------------------
__global__ void __launch_bounds__(TPB)
k_cand(const float* __restrict__ x, const float* __restrict__ H,
       const float* __restrict__ ax, const float* __restrict__ aHR,
       const float* __restrict__ HR,
       const float* __restrict__ Wxh, const float* __restrict__ bxh,
       const float* __restrict__ Whh, const float* __restrict__ bhh,
       const float* __restrict__ Z, float* __restrict__ h_out, int N) {
  __shared__ __align__(16) float sWxh[2 * WSZ], sWhh[2 * WSZ];
  __shared__ float sbh[F];
  stage_w(sWxh, Wxh);
  stage_w(sWhh, Whh);
  if (threadIdx.x < F) sbh[threadIdx.x] = bxh[threadIdx.x] + bhh[threadIdx.x];
  __syncthreads();

  int lane = threadIdx.x & 31;
  int tile = blockIdx.x * WAVES_PB + (threadIdx.x >> 5);
  int row0 = tile * 16;
  if (row0 >= N) return;

  int nlo = lane & 15;
  float bh0 = sbh[nlo], bh1 = sbh[16 + nlo];
  v8f c0 = {bh0, bh0, bh0, bh0, bh0, bh0, bh0, bh0};
  v8f c1 = {bh1, bh1, bh1, bh1, bh1, bh1, bh1, bh1};

#pragma unroll
  for (int k0 = 0; k0 < F; k0 += 4) {
    v2f fx   = afrag(x,   row0, lane, k0, N);
    v2f fax  = afrag(ax,  row0, lane, k0, N);
    v2f fhr  = afrag(HR,  row0, lane, k0, N);
    v2f fahr = afrag(aHR, row0, lane, k0, N);
    c0 = wmma4(fx,   bfragS(sWxh,       lane, k0, 0),  c0);
    c1 = wmma4(fx,   bfragS(sWxh,       lane, k0, 16), c1);
    c0 = wmma4(fax,  bfragS(sWxh + WSZ, lane, k0, 0),  c0);
    c1 = wmma4(fax,  bfragS(sWxh + WSZ, lane, k0, 16), c1);
    c0 = wmma4(fhr,  bfragS(sWhh,       lane, k0, 0),  c0);
    c1 = wmma4(fhr,  bfragS(sWhh,       lane, k0, 16), c1);
    c0 = wmma4(fahr, bfragS(sWhh + WSZ, lane, k0, 0),  c0);
    c1 = wmma4(fahr, bfragS(sWhh + WSZ, lane, k0, 16), c1);
  }

  int rbase = row0 + ((lane >> 4) << 3);
#pragma unroll
  for (int r = 0; r < 8; ++r) {
    int m = rbase + r;
    if (m >= N) break;
    size_t o0 = (size_t)m * F + nlo, o1 = o0 + 16;
    float ht0 = tanhf(c0[r]), ht1 = tanhf(c1[r]);
    float zv0 = Z[o0], zv1 = Z[o1];
    h_out[o0] = zv0 * H[o0] + (1.0f - zv0) * ht0;
    h_out[o1] = zv1 * H[o1] + (1.0f - zv1) * ht1;
  }
}

// out = relu(h) @ lin_w(32x4) + lin_b
__global__ void k_out(const float* __restrict__ h,
                      const float* __restrict__ lin_w,
                      const float* __restrict__ lin_b,
                      float* __restrict__ out, int N) {
  int i = blockIdx.x * blockDim.x + threadIdx.x;
  if (i >= N) return;
  float acc0 = lin_b[0], acc1 = lin_b[1], acc2 = lin_b[2], acc3 = lin_b[3];
  const float* hr = h + (size_t)i * F;
#pragma unroll
  for (int f = 0; f < F; ++f) {
    float v = fmaxf(hr[f], 0.0f);
    acc0 += v * lin_w[f * 4 + 0];
    acc1 += v * lin_w[f * 4 + 1];
    acc2 += v * lin_w[f * 4 + 2];
    acc3 += v * lin_w[f * 4 + 3];
  }
  float* o = out + (size_t)i * 4;
  o[0] = acc0; o[1] = acc1; o[2] = acc2; o[3] = acc3;
}

// ---------------------------------------------------------------------------
extern "C" void kernel_launch(void* const* d_in, const int* in_sizes, int n_in,
                              void* d_out, int out_size, void* d_ws, size_t ws_size,
                              hipStream_t stream) {
  const float*     x    = (const float*)d_in[0];
  const long long* ei   = (const long long*)d_in[1];
  const float*     ew   = (const float*)d_in[2];
  const float*     H    = (const float*)d_in[3];
  const float* Wxz = (const float*)d_in[4];  const float* bxz = (const float*)d_in[5];
  const float* Whz = (const float*)d_in[6];  const float* bhz = (const float*)d_in[7];
  const float* Wxr = (const float*)d_in[8];  const float* bxr = (const float*)d_in[9];
  const float* Whr = (const float*)d_in[10]; const float* bhr = (const float*)d_in[11];
  const float* Wxh = (const float*)d_in[12]; const float* bxh = (const float*)d_in[13];
  const float* Whh = (const float*)d_in[14]; const float* bhh = (const float*)d_in[15];
  const float* lin_w = (const float*)d_in[16];
  const float* lin_b = (const float*)d_in[17];

  const int N = in_sizes[0] / F;
  const int E = in_sizes[2];

  // workspace layout (floats): [deg N][aggx 32N][aggH 32N][aggHR 32N][nw E][Z 32N][HR 32N]
  float* ws    = (float*)d_ws;
  float* deg   = ws;                       // also dinv (in place)
  float* aggx  = deg   + N;
  float* aggH  = aggx  + (size_t)F * N;
  float* aggHR = aggH  + (size_t)F * N;
  float* nw    = aggHR + (size_t)F * N;
  float* Zbuf  = nw    + E;
  float* HRbuf = Zbuf  + (size_t)F * N;

  float* out_p = (float*)d_out;            // (N,4)
  float* h_out = out_p + (size_t)N * 4;    // (N,32)

  const int gE  = (E + TPB - 1) / TPB;               // thread-per-edge
  const int gEw = (E + WAVES_PB - 1) / WAVES_PB;     // wave-per-edge
  const int gN  = (N + TPB - 1) / TPB;
  const int tiles = (N + 15) / 16;
  const int gT  = (tiles + WAVES_PB - 1) / WAVES_PB;

  k_zero  <<<4096, TPB, 0, stream>>>(deg, (long long)(1 + 3 * F) * N);
  k_degree<<<gE,   TPB, 0, stream>>>(ei, ew, deg, E);
  k_dinv  <<<gN,   TPB, 0, stream>>>(deg, N);
  k_nw    <<<gE,   TPB, 0, stream>>>(ei, ew, deg, nw, E);
  k_spmm2 <<<gEw,  TPB, 0, stream>>>(ei, nw, x, H, aggx, aggH, E);
  k_gate  <<<gT,   TPB, 0, stream>>>(x, H, aggx, aggH,
                                     Wxz, bxz, Whz, bhz, Wxr, bxr, Whr, bhr,
                                     Zbuf, HRbuf, N);
  k_spmm1 <<<gEw,  TPB, 0, stream>>>(ei, nw, HRbuf, aggHR, E);
  k_cand  <<<gT,   TPB, 0, stream>>>(x, H, aggx, aggHR, HRbuf,
                                     Wxh, bxh, Whh, bhh, Zbuf, h_out, N);
  k_out   <<<gN,   TPB, 0, stream>>>(h_out, lin_w, lin_b, out_p, N);

  (void)n_in; (void)out_size; (void)ws_size;
}